// SelfAttention_55027120996456
// MI455X (gfx1250) — compile-verified
//
#include <hip/hip_runtime.h>

// MI455X / gfx1250, wave32. All matmuls use V_WMMA_F32_16X16X4_F32 (fp32 in,
// fp32 accumulate). The op is HBM-bound (~137 GFLOP vs ~350MB of traffic at
// 23.3 TB/s => ~15us floor), so fp32 WMMA preserves reference precision at
// zero performance cost versus downcasting. A-tile staging uses the CDNA5
// async global->LDS DMA path (GLOBAL_LOAD_ASYNC_TO_LDS_B128, ASYNCcnt);
// B tiles are staged k-pair-interleaved so every WMMA B fragment is a single
// 8-byte conflict-free LDS load (no v_mov shuffling in the hot loop).

typedef float v2f __attribute__((ext_vector_type(2)));
typedef float v8f __attribute__((ext_vector_type(8)));

#define WMMA_F32(a, b, c) \
  __builtin_amdgcn_wmma_f32_16x16x4_f32(false, (a), false, (b), (short)0, (c), false, false)

static constexpr int Dm = 1024;   // model dim
static constexpr int Bb = 4;      // batch
static constexpr int Ss = 2048;   // seq len
#define NEG_INF (-__builtin_inff())

// CDNA5 async global->LDS copy, 16B per lane. LDS address comes from a VGPR;
// the LDS aperture maps generic addr[31:0] to the LDS offset, so the low
// 32 bits of a generic pointer to __shared__ are the LDS address VGPR.
__device__ __forceinline__ void async_copy_b128(const void* gsrc, void* ldst) {
  unsigned loff = (unsigned)(unsigned long long)ldst;
  asm volatile("global_load_async_to_lds_b128 %0, %1, off"
               :: "v"(loff), "v"(gsrc) : "memory");
}
__device__ __forceinline__ void wait_async0() {
  asm volatile("s_wait_asynccnt 0x0" ::: "memory");
}

// ---------------------------------------------------------------------------
// Generic fp32 GEMM + optional bias: C[M,N] = A[M,K] @ W[K,N] + bias[N]
// Block tile 128x64, BK=32. A tile async-staged row-major; B tile staged
// k-pair interleaved: Bs2[kk][n] = (W[2kk][n], W[2kk+1][n]) so a WMMA B
// fragment is one ds_load_b64. 8 waves, each 16(M) x 64(N): one A fragment
// feeds 4 chained wmma_f32_16x16x4 per k-step.
// Requires M%128==0, N%64==0, K%32==0 (true for all uses here).
// ---------------------------------------------------------------------------
__global__ __launch_bounds__(256)
void gemm_bias_kernel(const float* __restrict__ A, int lda,
                      const float* __restrict__ W, int ldb,
                      const float* __restrict__ bias,
                      float* __restrict__ C, int ldc,
                      int M, int N, int K)
{
  constexpr int BM = 128, BN = 64, BK = 32;
  constexpr int LDAS = BK + 4;      // 36 floats: 8B-aligned, conflict-free
  constexpr int LDBS2 = BN + 16;    // 80 float2 = 160 words; 160%64=32 ->
                                    // the two half-waves use disjoint banks
  __shared__ float As[BM * LDAS];
  __shared__ float Bs[(BK / 2) * LDBS2 * 2];   // 16 rows of 80 float2

  const int tid  = threadIdx.x;
  const int lane = tid & 31;
  const int wave = tid >> 5;      // 0..7 -> M sub-tile (16 rows each)
  const int hl   = lane >> 4;     // half-wave: K pair 0/1 vs 2/3
  const int l15  = lane & 15;
  const int bm   = blockIdx.y * BM;
  const int bn   = blockIdx.x * BN;

  v8f c0 = {}, c1 = {}, c2 = {}, c3 = {};   // N = [0,16),[16,32),[32,48),[48,64)

  for (int kb = 0; kb < K; kb += BK) {
    // A tile 128x32: async DMA, 1024 16B chunks over 256 threads
#pragma unroll
    for (int i = 0; i < 4; ++i) {
      int idx = tid + i * 256;                 // 0..1023
      int ra  = idx >> 3;                      // 0..127
      int ca  = (idx & 7) << 2;                // 0..28
      async_copy_b128(&A[(size_t)(bm + ra) * lda + kb + ca], &As[ra * LDAS + ca]);
    }
    // B tile 32x64 -> interleaved: thread grabs float2 from rows 2kk,2kk+1
    // (coalesced within each row) and writes one b128 covering Bs2[kk][n..n+1]
#pragma unroll
    for (int i = 0; i < 2; ++i) {
      int idx = tid + i * 256;                 // 0..511
      int kk  = idx >> 5;                      // 0..15
      int n   = (idx & 31) << 1;               // 0..62
      v2f g0 = *reinterpret_cast<const v2f*>(&W[(size_t)(kb + 2 * kk) * ldb + bn + n]);
      v2f g1 = *reinterpret_cast<const v2f*>(&W[(size_t)(kb + 2 * kk + 1) * ldb + bn + n]);
      float4 st; st.x = g0.x; st.y = g1.x; st.z = g0.y; st.w = g1.y;
      *reinterpret_cast<float4*>(&Bs[(kk * LDBS2 + n) * 2]) = st;
    }
    wait_async0();
    __syncthreads();

#pragma unroll
    for (int k4 = 0; k4 < BK; k4 += 4) {
      const int ks = k4 + 2 * hl;
      const int kk = (k4 >> 1) + hl;           // interleaved B row
      // A frag: lanes 0-15 rows M=lane with K pair (k4,k4+1); lanes 16-31 (k4+2,k4+3)
      v2f a = *reinterpret_cast<const v2f*>(&As[(wave * 16 + l15) * LDAS + ks]);
      const float* bp = &Bs[(kk * LDBS2 + l15) * 2];
      v2f b0 = *reinterpret_cast<const v2f*>(bp);        // N block 0
      v2f b1 = *reinterpret_cast<const v2f*>(bp + 32);   // +16 float2
      v2f b2 = *reinterpret_cast<const v2f*>(bp + 64);
      v2f b3 = *reinterpret_cast<const v2f*>(bp + 96);
      c0 = WMMA_F32(a, b0, c0);
      c1 = WMMA_F32(a, b1, c1);
      c2 = WMMA_F32(a, b2, c2);
      c3 = WMMA_F32(a, b3, c3);
    }
    __syncthreads();
  }

  // C/D layout: VGPR r -> rows {r, r+8}; lanes of each half hold N=0..15
  const int col = bn + l15;
  const float bia0 = bias ? bias[col]      : 0.f;
  const float bia1 = bias ? bias[col + 16] : 0.f;
  const float bia2 = bias ? bias[col + 32] : 0.f;
  const float bia3 = bias ? bias[col + 48] : 0.f;
#pragma unroll
  for (int r = 0; r < 8; ++r) {
    const int row = bm + wave * 16 + r + hl * 8;
    float* crow = &C[(size_t)row * ldc];
    crow[col]      = c0[r] + bia0;
    crow[col + 16] = c1[r] + bia1;
    crow[col + 32] = c2[r] + bia2;
    crow[col + 48] = c3[r] + bia3;
  }
}

// ---------------------------------------------------------------------------
// Scores + online softmax stats. One block = (batch b, 16 query rows).
// Q tile (16 x 1024) async-staged to LDS (stride 1028 -> 8B-aligned,
// conflict-free half-wave frag reads). 4 waves stride over 16-wide key tiles;
// each 16x16 score tile = 256 x wmma_f32_16x16x4 over D. Raw masked/scaled
// scores -> P; flash-style running (rowmax, rowsum) via half-wave shuffles
// (wave32), combined across waves in LDS.
// ---------------------------------------------------------------------------
__global__ __launch_bounds__(128)
void scores_kernel(const float* __restrict__ Q, const float* __restrict__ Km,
                   const int* __restrict__ mask,
                   float* __restrict__ P, float* __restrict__ gm, float* __restrict__ gz)
{
  constexpr int QLD = Dm + 4;                 // 1028
  __shared__ float Qs[16 * QLD];
  __shared__ float smx[4 * 16];
  __shared__ float ssz[4 * 16];

  const int b    = blockIdx.y;
  const int q0   = blockIdx.x * 16;
  const int tid  = threadIdx.x;
  const int lane = tid & 31;
  const int wave = tid >> 5;                  // 0..3
  const int hl   = lane >> 4;
  const int l15  = lane & 15;

  const float* Qb = Q  + (size_t)b * Ss * Dm;
  const float* Kb = Km + (size_t)b * Ss * Dm;
  float*       Pb = P  + (size_t)b * Ss * Ss;

  // Async-stage 16x1024 Q tile: 4096 16B chunks over 128 threads
#pragma unroll
  for (int i = 0; i < 32; ++i) {
    int idx = tid + i * 128;
    int r = idx >> 8;                         // 0..15
    int c = (idx & 255) << 2;                 // 0..1020
    async_copy_b128(&Qb[(size_t)(q0 + r) * Dm + c], &Qs[r * QLD + c]);
  }
  wait_async0();
  __syncthreads();

  float m[8], z[8];
#pragma unroll
  for (int r = 0; r < 8; ++r) { m[r] = NEG_INF; z[r] = 0.f; }

  const float scale = 0.03125f;               // 1/sqrt(1024)

  for (int kt = wave; kt < Ss / 16; kt += 4) {
    const int kbase = kt * 16;
    v8f acc = {};
    const float* qrow = &Qs[l15 * QLD + 2 * hl];
    const float* krow = &Kb[(size_t)(kbase + l15) * Dm + 2 * hl];
#pragma unroll 8
    for (int d = 0; d < Dm; d += 4) {
      v2f a  = *reinterpret_cast<const v2f*>(qrow + d);           // Q frag (A)
      v2f bb = *reinterpret_cast<const v2f*>(krow + d);           // K^T frag (B)
      acc = WMMA_F32(a, bb, acc);
    }
#pragma unroll
    for (int r = 0; r < 8; ++r) {
      const int qi = q0 + r + hl * 8;
      const int ki = kbase + l15;
      float s = acc[r] * scale;
      if (mask[((size_t)b * Ss + qi) * Ss + ki] == 0) s = NEG_INF;
      Pb[(size_t)qi * Ss + ki] = s;           // raw masked score
      float tm = s;
#pragma unroll
      for (int off = 8; off; off >>= 1) tm = fmaxf(tm, __shfl_xor(tm, off, 16));
      const float mn = fmaxf(m[r], tm);
      float p = (s == NEG_INF) ? 0.f : __expf(s - mn);
#pragma unroll
      for (int off = 8; off; off >>= 1) p += __shfl_xor(p, off, 16);
      const float resc = (m[r] == mn) ? 1.f : __expf(m[r] - mn);
      z[r] = z[r] * resc + p;
      m[r] = mn;
    }
  }

  // combine the 4 waves' partial stats
  if (l15 == 0) {
#pragma unroll
    for (int r = 0; r < 8; ++r) {
      smx[wave * 16 + hl * 8 + r] = m[r];
      ssz[wave * 16 + hl * 8 + r] = z[r];
    }
  }
  __syncthreads();
  if (tid < 16) {
    float M = NEG_INF;
    for (int w = 0; w < 4; ++w) M = fmaxf(M, smx[w * 16 + tid]);
    float Z = 0.f;
    for (int w = 0; w < 4; ++w) {
      float zw = ssz[w * 16 + tid];
      if (zw != 0.f) Z += zw * __expf(smx[w * 16 + tid] - M);
    }
    gm[(size_t)b * Ss + q0 + tid] = M;
    gz[(size_t)b * Ss + q0 + tid] = Z;
  }
}

// ---------------------------------------------------------------------------
// P[i] = exp(P[i] - m_row) / z_row   (elementwise, streams 64MB r+w)
// ---------------------------------------------------------------------------
__global__ __launch_bounds__(256)
void softmax_norm_kernel(float* __restrict__ P, const float* __restrict__ gm,
                         const float* __restrict__ gz, size_t total)
{
  size_t i = (size_t)blockIdx.x * 256 + threadIdx.x;
  if (i >= total) return;
  size_t row = i >> 11;                       // / Ss (2048) -> b*S + q
  float s = P[i];
  P[i] = (s == NEG_INF) ? 0.f : __expf(s - gm[row]) / gz[row];
}

// ---------------------------------------------------------------------------
extern "C" void kernel_launch(void* const* d_in, const int* in_sizes, int n_in,
                              void* d_out, int out_size, void* d_ws, size_t ws_size,
                              hipStream_t stream)
{
  (void)in_sizes; (void)n_in; (void)out_size; (void)ws_size;
  const float* X  = (const float*)d_in[0];
  const int*   am = (const int*)  d_in[1];
  const float* Wq = (const float*)d_in[2];
  const float* bq = (const float*)d_in[3];
  const float* Wk = (const float*)d_in[4];
  const float* bk = (const float*)d_in[5];
  const float* Wv = (const float*)d_in[6];
  const float* bv = (const float*)d_in[7];
  const float* Wo = (const float*)d_in[8];
  const float* bo = (const float*)d_in[9];
  float* out = (float*)d_out;

  const size_t NTOK = (size_t)Bb * Ss;        // 8192 rows
  // workspace layout (floats): Q | K | V | attn_out | P | gm | gz  (~192 MB)
  float* ws = (float*)d_ws;
  float* Qb = ws;
  float* Kb = Qb + NTOK * Dm;
  float* Vb = Kb + NTOK * Dm;
  float* AO = Vb + NTOK * Dm;
  float* P  = AO + NTOK * Dm;
  float* gm = P + (size_t)Bb * Ss * Ss;
  float* gz = gm + NTOK;

  const dim3 blk(256);
  const dim3 gproj(Dm / 64, (unsigned)(NTOK / 128));      // 16 x 64

  // QKV projections: [8192,1024] @ [1024,1024] + bias
  gemm_bias_kernel<<<gproj, blk, 0, stream>>>(X, Dm, Wq, Dm, bq, Qb, Dm, (int)NTOK, Dm, Dm);
  gemm_bias_kernel<<<gproj, blk, 0, stream>>>(X, Dm, Wk, Dm, bk, Kb, Dm, (int)NTOK, Dm, Dm);
  gemm_bias_kernel<<<gproj, blk, 0, stream>>>(X, Dm, Wv, Dm, bv, Vb, Dm, (int)NTOK, Dm, Dm);

  // scores + mask + online softmax stats
  scores_kernel<<<dim3(Ss / 16, Bb), dim3(128), 0, stream>>>(Qb, Kb, am, P, gm, gz);

  // normalize probabilities
  const size_t total = (size_t)Bb * Ss * Ss;
  softmax_norm_kernel<<<dim3((unsigned)((total + 255) / 256)), blk, 0, stream>>>(P, gm, gz, total);

  // attn_out = P @ V per batch: [2048,2048] @ [2048,1024]
  const dim3 gpv(Dm / 64, Ss / 128);                      // 16 x 16
  for (int b = 0; b < Bb; ++b) {
    gemm_bias_kernel<<<gpv, blk, 0, stream>>>(P + (size_t)b * Ss * Ss, Ss,
                                              Vb + (size_t)b * Ss * Dm, Dm, nullptr,
                                              AO + (size_t)b * Ss * Dm, Dm, Ss, Dm, Ss);
  }

  // output projection: [8192,1024] @ [1024,1024] + bias
  gemm_bias_kernel<<<gproj, blk, 0, stream>>>(AO, Dm, Wo, Dm, bo, out, Dm, (int)NTOK, Dm, Dm);
}